// GAT_GCN_4604204941354
// MI455X (gfx1250) — compile-verified
//
#include <hip/hip_runtime.h>
#include <hip/hip_bf16.h>
#include <cstdint>

// ---------------------------------------------------------------------------
// MI455X (gfx1250) GAT+GCN pipeline.
// f32 GEMMs via V_WMMA_F32_16X16X4_F32 (matches f32 reference numerics).
// Block tile 64x128, 8 waves (4Mx2N), wave tile 16x64 = 4 accumulators,
// LDS staged via GLOBAL_LOAD_ASYNC_TO_LDS_B128 (ASYNCcnt path).
// Edge ops are L2-resident streaming + f32/u32 atomics.
// ---------------------------------------------------------------------------

typedef float v2f __attribute__((ext_vector_type(2)));
typedef float v8f __attribute__((ext_vector_type(8)));

#if defined(__has_builtin)
#  if __has_builtin(__builtin_amdgcn_global_load_async_to_lds_b128) && \
      __has_builtin(__builtin_amdgcn_s_wait_asynccnt)
#    define HAVE_ASYNC_LDS 1
#  endif
#endif

#if HAVE_ASYNC_LDS
typedef int i32x4 __attribute__((vector_size(16)));
typedef __attribute__((address_space(1))) i32x4 gls_i32x4;
typedef __attribute__((address_space(3))) i32x4 lds_i32x4;
#define ASYNC_COPY_B128(gsrc, ldst)                                           \
  __builtin_amdgcn_global_load_async_to_lds_b128(                             \
      (gls_i32x4*)(uintptr_t)(gsrc), (lds_i32x4*)(uintptr_t)(ldst), 0, 0)
#define ASYNC_WAIT() __builtin_amdgcn_s_wait_asynccnt(0)
#else
#define ASYNC_COPY_B128(gsrc, ldst)                                           \
  do {                                                                        \
    const float* _g = (const float*)(gsrc);                                   \
    float* _l = (float*)(ldst);                                               \
    _l[0] = _g[0]; _l[1] = _g[1]; _l[2] = _g[2]; _l[3] = _g[3];               \
  } while (0)
#define ASYNC_WAIT() ((void)0)
#endif

#define ENC_NEG_INF 0x007FFFFFu  // enc(-inf) for ordered-uint float max

__device__ __forceinline__ unsigned encf(float f) {
  unsigned u = __float_as_uint(f);
  return (u & 0x80000000u) ? ~u : (u | 0x80000000u);
}
__device__ __forceinline__ float decf(unsigned u) {
  u = (u & 0x80000000u) ? (u & 0x7FFFFFFFu) : ~u;
  return __uint_as_float(u);
}

// ---------------------------------------------------------------------------
// f32 WMMA GEMM:  C[M,N] = act(A[M,K] @ B[K,N] + bias)
// Block: 256 threads = 8 waves; block tile 64(M) x 128(N); K-step 16.
// Wave (wm 0..3, wn 0..1) owns 16x64 strip = four 16x16 WMMA tiles.
// LDS strides: bank-conflict-free reads AND 16B-aligned b128 stores:
//   As stride 20 floats (gcd(20,64)=4 -> 16 distinct banks; 80B rows)
//   Bs stride 136 floats (2*136 mod 64 = 16 -> half-waves bank-disjoint)
// ---------------------------------------------------------------------------
__global__ __launch_bounds__(256)
void k_wmma_gemm(const float* __restrict__ A, const float* __restrict__ B,
                 const float* __restrict__ bias, float* __restrict__ C,
                 int M, int N, int K, int act) {
  __shared__ float As[64 * 20];
  __shared__ float Bs[16 * 136];

  const int tid   = threadIdx.x;
  const int lane  = tid & 31;
  const int wave  = tid >> 5;
  const int wm    = wave & 3;
  const int wn    = wave >> 2;
  const int khalf = lane >> 4;   // 0: K={k,k+1}; 1: K={k+2,k+3}
  const int l16   = lane & 15;
  const int m0    = blockIdx.y * 64;
  const int n0    = blockIdx.x * 128;

  v8f acc[4] = {v8f{}, v8f{}, v8f{}, v8f{}};

  // A staging: thread t -> row = t/4, 4-float segment = (t%4)*4  (one b128)
  const int ar = tid >> 2;
  const int ac = (tid & 3) * 4;

  for (int k0 = 0; k0 < K; k0 += 16) {
    {
      const int gr = m0 + ar;
      float* ldst = &As[ar * 20 + ac];
      if (gr < M) {
        ASYNC_COPY_B128(A + (size_t)gr * K + k0 + ac, ldst);
      } else {
        ldst[0] = 0.0f; ldst[1] = 0.0f; ldst[2] = 0.0f; ldst[3] = 0.0f;
      }
    }
    // B staging: 2 b128 segments per thread; seg s -> row = s/32, col = (s%32)*4
#pragma unroll
    for (int s = 0; s < 2; ++s) {
      const int seg = tid + s * 256;
      const int br  = seg >> 5;
      const int bc  = (seg & 31) * 4;
      const int gc  = n0 + bc;
      float* ldst = &Bs[br * 136 + bc];
      if (gc + 3 < N) {
        ASYNC_COPY_B128(B + (size_t)(k0 + br) * N + gc, ldst);
      } else {
#pragma unroll
        for (int j = 0; j < 4; ++j)
          ldst[j] = (gc + j < N) ? B[(size_t)(k0 + br) * N + gc + j] : 0.0f;
      }
    }
    ASYNC_WAIT();
    __syncthreads();

#pragma unroll
    for (int ks = 0; ks < 4; ++ks) {
      const int ka = ks * 4 + khalf * 2;
      v2f a;
      a.x = As[(wm * 16 + l16) * 20 + ka];
      a.y = As[(wm * 16 + l16) * 20 + ka + 1];
#pragma unroll
      for (int nt = 0; nt < 4; ++nt) {
        const int nc = wn * 64 + nt * 16 + l16;
        v2f b;
        b.x = Bs[ka * 136 + nc];
        b.y = Bs[(ka + 1) * 136 + nc];
        acc[nt] = __builtin_amdgcn_wmma_f32_16x16x4_f32(false, a, false, b,
                                                        (short)0, acc[nt],
                                                        false, false);
      }
    }
    __syncthreads();
  }

  // C/D layout: VGPR v -> row M = v + 8*khalf ; lane%16 -> col N
  const int row_base = m0 + wm * 16 + khalf * 8;
#pragma unroll
  for (int nt = 0; nt < 4; ++nt) {
    const int col = n0 + wn * 64 + nt * 16 + l16;
    if (col < N) {
      const float bv = bias ? bias[col] : 0.0f;
#pragma unroll
      for (int v = 0; v < 8; ++v) {
        const int row = row_base + v;
        if (row < M) {
          float xv = acc[nt][v] + bv;
          if (act) xv = fmaxf(xv, 0.0f);
          C[(size_t)row * N + col] = xv;
        }
      }
    }
  }
}

// ---------------------------------------------------------------------------
// Elementwise helpers
// ---------------------------------------------------------------------------
__global__ void k_fill_f32(float* p, float v, long n) {
  long i = (long)blockIdx.x * blockDim.x + threadIdx.x;
  if (i < n) p[i] = v;
}
__global__ void k_fill_u32(unsigned* p, unsigned v, long n) {
  long i = (long)blockIdx.x * blockDim.x + threadIdx.x;
  if (i < n) p[i] = v;
}
__global__ void k_init_rows(float* p, const float* __restrict__ bias, int rows, int cols) {
  long i = (long)blockIdx.x * blockDim.x + threadIdx.x;
  if (i < (long)rows * cols) p[i] = bias[i % cols];
}
__global__ void k_concat(const float* __restrict__ x, const float* __restrict__ pe,
                         float* __restrict__ xcat, int Nn, int F, int PEd) {
  const int Fin = F + PEd;
  long i = (long)blockIdx.x * blockDim.x + threadIdx.x;
  if (i >= (long)Nn * Fin) return;
  int r = (int)(i / Fin), c = (int)(i % Fin);
  xcat[i] = (c < F) ? x[(size_t)r * F + c] : pe[(size_t)r * PEd + (c - F)];
}
__global__ void k_bias_relu(float* p, const float* __restrict__ bias, int rows, int cols) {
  long i = (long)blockIdx.x * blockDim.x + threadIdx.x;
  if (i < (long)rows * cols) p[i] = fmaxf(p[i] + bias[i % cols], 0.0f);
}

// ---------------------------------------------------------------------------
// GATv2 edge scoring: one wave per edge (self-loops appended at e>=E)
// ---------------------------------------------------------------------------
__global__ __launch_bounds__(256)
void k_edge_score(const float* __restrict__ hl, const float* __restrict__ hr,
                  const float* __restrict__ att, const int* __restrict__ ei,
                  float* __restrict__ score, unsigned* __restrict__ smaxu,
                  float* __restrict__ deg, int E, int Ep, int H, int C) {
  const int wid  = (blockIdx.x * blockDim.x + threadIdx.x) >> 5;
  const int lane = threadIdx.x & 31;
  if (wid >= Ep) return;
  int src, dst;
  if (wid < E) { src = ei[wid]; dst = ei[E + wid]; }
  else         { src = dst = wid - E; }
  const int HC = H * C;
  const float* pl = hl + (size_t)src * HC;
  const float* pr = hr + (size_t)dst * HC;
  if (lane == 0) atomicAdd(deg + dst, 1.0f);
  for (int h = 0; h < H; ++h) {
    float s = 0.0f;
    for (int c = lane; c < C; c += 32) {
      float v = pl[h * C + c] + pr[h * C + c];
      v = (v > 0.0f) ? v : 0.2f * v;  // leaky_relu(0.2)
      s += v * att[h * C + c];
    }
    for (int off = 16; off; off >>= 1) s += __shfl_xor(s, off, 32);
    if (lane == 0) {
      score[(size_t)wid * H + h] = s;
      atomicMax(smaxu + (size_t)dst * H + h, encf(s));
    }
  }
}

// a = exp(score - smax[dst]); accumulate segment sum
__global__ void k_edge_softnorm(const int* __restrict__ ei, float* __restrict__ score,
                                const unsigned* __restrict__ smaxu, float* __restrict__ ssum,
                                int E, int Ep, int H) {
  long i = (long)blockIdx.x * blockDim.x + threadIdx.x;
  if (i >= (long)Ep * H) return;
  int e = (int)(i / H), h = (int)(i % H);
  int dst = (e < E) ? ei[E + e] : (e - E);
  float a = expf(score[i] - decf(smaxu[(size_t)dst * H + h]));
  score[i] = a;
  atomicAdd(ssum + (size_t)dst * H + h, a);
}

// x1[dst] += (a/ssum[dst]) * hl[src]  — one wave per edge
__global__ __launch_bounds__(256)
void k_gat_aggregate(const float* __restrict__ hl, const int* __restrict__ ei,
                     const float* __restrict__ score, const float* __restrict__ ssum,
                     float* __restrict__ x1, int E, int Ep, int H, int C) {
  const int wid  = (blockIdx.x * blockDim.x + threadIdx.x) >> 5;
  const int lane = threadIdx.x & 31;
  if (wid >= Ep) return;
  int src, dst;
  if (wid < E) { src = ei[wid]; dst = ei[E + wid]; }
  else         { src = dst = wid - E; }
  const int HC = H * C;
  float wv = 0.0f;
  if (lane < H) wv = score[(size_t)wid * H + lane] / ssum[(size_t)dst * H + lane];
  const float* pl = hl + (size_t)src * HC;
  float* po = x1 + (size_t)dst * HC;
  for (int c = lane; c < HC; c += 32) {
    float w = __shfl(wv, c / C, 32);
    atomicAdd(po + c, w * pl[c]);
  }
}

// x2[dst] += dinv[src]*dinv[dst] * xw[src] — one wave per edge
__global__ __launch_bounds__(256)
void k_gcn_aggregate(const float* __restrict__ xw, const int* __restrict__ ei,
                     const float* __restrict__ deg, float* __restrict__ x2,
                     int E, int Ep, int HC) {
  const int wid  = (blockIdx.x * blockDim.x + threadIdx.x) >> 5;
  const int lane = threadIdx.x & 31;
  if (wid >= Ep) return;
  int src, dst;
  if (wid < E) { src = ei[wid]; dst = ei[E + wid]; }
  else         { src = dst = wid - E; }
  float nrm = 0.0f;
  if (lane == 0) {
    float a = deg[src], b = deg[dst];
    nrm = ((a > 0.0f) ? rsqrtf(fmaxf(a, 1.0f)) : 0.0f) *
          ((b > 0.0f) ? rsqrtf(fmaxf(b, 1.0f)) : 0.0f);
  }
  nrm = __shfl(nrm, 0, 32);
  const float* ps = xw + (size_t)src * HC;
  float* po = x2 + (size_t)dst * HC;
  for (int c = lane; c < HC; c += 32) atomicAdd(po + c, nrm * ps[c]);
}

// Pool accumulation: one wave per node
__global__ __launch_bounds__(256)
void k_pool_accum(const float* __restrict__ x2, const int* __restrict__ batch,
                  float* __restrict__ pm, unsigned* __restrict__ pxu,
                  float* __restrict__ cnt, int Nn, int HC) {
  const int wid  = (blockIdx.x * blockDim.x + threadIdx.x) >> 5;
  const int lane = threadIdx.x & 31;
  if (wid >= Nn) return;
  const int g = batch[wid];
  if (lane == 0) atomicAdd(cnt + g, 1.0f);
  const float* pr = x2 + (size_t)wid * HC;
  for (int c = lane; c < HC; c += 32) {
    float v = pr[c];
    atomicAdd(pm + (size_t)g * HC + c, v);
    atomicMax(pxu + (size_t)g * HC + c, encf(v));
  }
}

// h[g] = concat(mean, max)
__global__ void k_pool_final(const float* __restrict__ pm, const unsigned* __restrict__ pxu,
                             const float* __restrict__ cnt, float* __restrict__ h,
                             int G, int HC) {
  long i = (long)blockIdx.x * blockDim.x + threadIdx.x;
  if (i >= (long)G * 2 * HC) return;
  int g = (int)(i / (2 * HC)), j = (int)(i % (2 * HC));
  float c = cnt[g];
  if (j < HC) h[i] = pm[(size_t)g * HC + j] / fmaxf(c, 1.0f);
  else        h[i] = (c > 0.0f) ? decf(pxu[(size_t)g * HC + (j - HC)]) : 0.0f;
}

// out[g] = h3[g,:] . wfin — one wave per graph
__global__ __launch_bounds__(256)
void k_final_dot(const float* __restrict__ h3, const float* __restrict__ wfin,
                 float* __restrict__ out, int G, int K) {
  const int wid  = (blockIdx.x * blockDim.x + threadIdx.x) >> 5;
  const int lane = threadIdx.x & 31;
  if (wid >= G) return;
  float s = 0.0f;
  for (int k = lane; k < K; k += 32) s += h3[(size_t)wid * K + k] * wfin[k];
  for (int off = 16; off; off >>= 1) s += __shfl_xor(s, off, 32);
  if (lane == 0) out[wid] = s;
}

// ---------------------------------------------------------------------------
static inline int cdiv(long a, long b) { return (int)((a + b - 1) / b); }

extern "C" void kernel_launch(void* const* d_in, const int* in_sizes, int n_in,
                              void* d_out, int out_size, void* d_ws, size_t ws_size,
                              hipStream_t stream) {
  const float* x      = (const float*)d_in[0];
  const float* pe     = (const float*)d_in[1];
  const int*   ei     = (const int*)d_in[2];
  const int*   batch  = (const int*)d_in[3];
  const float* Wl     = (const float*)d_in[4];
  const float* bl     = (const float*)d_in[5];
  const float* Wr     = (const float*)d_in[6];
  const float* br     = (const float*)d_in[7];
  const float* att    = (const float*)d_in[8];
  const float* b_gat  = (const float*)d_in[9];
  const float* Wg     = (const float*)d_in[10];
  const float* bg     = (const float*)d_in[11];
  const float* W_fc1  = (const float*)d_in[12];
  const float* b_fc1  = (const float*)d_in[13];
  const float* W_go   = (const float*)d_in[14];
  const float* b_go   = (const float*)d_in[15];
  const float* W_semi = (const float*)d_in[16];
  const float* b_semi = (const float*)d_in[17];
  const float* W_fin  = (const float*)d_in[18];
  float* out = (float*)d_out;

  const int Nn   = in_sizes[3];          // 5000
  const int E    = in_sizes[2] / 2;      // 50000
  const int Ep   = E + Nn;               // + self-loops
  const int F    = in_sizes[0] / Nn;     // 768
  const int PEd  = in_sizes[1] / Nn;     // 32
  const int Fin  = F + PEd;              // 800
  const int HC   = in_sizes[5];          // 2400
  const int H    = 3;
  const int C    = HC / H;               // 800
  const int FC1  = in_sizes[13];         // 9600
  const int GO   = in_sizes[15];         // 800
  const int SEMI = in_sizes[17];         // 200
  const int G    = out_size;             // 32

  float* ws = (float*)d_ws;
  size_t o = 0;
  float*    xcat  = ws + o; o += (size_t)Nn * Fin;
  float*    hl    = ws + o; o += (size_t)Nn * HC;
  float*    hr    = ws + o; o += (size_t)Nn * HC;
  float*    x1    = ws + o; o += (size_t)Nn * HC;
  float*    xw    = ws + o; o += (size_t)Nn * HC;
  float*    x2    = ws + o; o += (size_t)Nn * HC;
  float*    score = ws + o; o += (size_t)Ep * H;
  unsigned* smaxu = (unsigned*)(ws + o); o += (size_t)Nn * H;
  float*    ssum  = ws + o; o += (size_t)Nn * H;
  float*    deg   = ws + o; o += (size_t)Nn;
  float*    pm    = ws + o; o += (size_t)G * HC;
  unsigned* pxu   = (unsigned*)(ws + o); o += (size_t)G * HC;
  float*    cnt   = ws + o; o += (size_t)G;
  float*    hcat  = ws + o; o += (size_t)G * 2 * HC;
  float*    h1    = ws + o; o += (size_t)G * FC1;
  float*    h2    = ws + o; o += (size_t)G * GO;
  float*    h3    = ws + o; o += (size_t)G * SEMI;
  (void)ws_size; (void)n_in;

  const int TB = 256;
  const int WPB = TB / 32;

  // 1) concat
  k_concat<<<cdiv((long)Nn * Fin, TB), TB, 0, stream>>>(x, pe, xcat, Nn, F, PEd);

  // 2) hl, hr GEMMs (WMMA)
  {
    dim3 g(cdiv(HC, 128), cdiv(Nn, 64));
    k_wmma_gemm<<<g, TB, 0, stream>>>(xcat, Wl, bl, hl, Nn, HC, Fin, 0);
    k_wmma_gemm<<<g, TB, 0, stream>>>(xcat, Wr, br, hr, Nn, HC, Fin, 0);
  }

  // 3) inits for scatter passes
  k_fill_f32<<<cdiv(Nn, TB), TB, 0, stream>>>(deg, 0.0f, Nn);
  k_fill_u32<<<cdiv((long)Nn * H, TB), TB, 0, stream>>>(smaxu, ENC_NEG_INF, (long)Nn * H);
  k_fill_f32<<<cdiv((long)Nn * H, TB), TB, 0, stream>>>(ssum, 0.0f, (long)Nn * H);
  k_fill_f32<<<cdiv((long)Nn * HC, TB), TB, 0, stream>>>(x1, 0.0f, (long)Nn * HC);
  k_init_rows<<<cdiv((long)Nn * HC, TB), TB, 0, stream>>>(x2, bg, Nn, HC);
  k_fill_f32<<<cdiv(G, TB), TB, 0, stream>>>(cnt, 0.0f, G);
  k_fill_f32<<<cdiv((long)G * HC, TB), TB, 0, stream>>>(pm, 0.0f, (long)G * HC);
  k_fill_u32<<<cdiv((long)G * HC, TB), TB, 0, stream>>>(pxu, ENC_NEG_INF, (long)G * HC);

  // 4) GATv2 edge pipeline
  k_edge_score<<<cdiv(Ep, WPB), TB, 0, stream>>>(hl, hr, att, ei, score, smaxu, deg, E, Ep, H, C);
  k_edge_softnorm<<<cdiv((long)Ep * H, TB), TB, 0, stream>>>(ei, score, smaxu, ssum, E, Ep, H);
  k_gat_aggregate<<<cdiv(Ep, WPB), TB, 0, stream>>>(hl, ei, score, ssum, x1, E, Ep, H, C);
  k_bias_relu<<<cdiv((long)Nn * HC, TB), TB, 0, stream>>>(x1, b_gat, Nn, HC);

  // 5) GCN: xw = x1 @ Wg ; scatter with sym-norm
  {
    dim3 g(cdiv(HC, 128), cdiv(Nn, 64));
    k_wmma_gemm<<<g, TB, 0, stream>>>(x1, Wg, nullptr, xw, Nn, HC, HC, 0);
  }
  k_gcn_aggregate<<<cdiv(Ep, WPB), TB, 0, stream>>>(xw, ei, deg, x2, E, Ep, HC);

  // 6) pooling
  k_pool_accum<<<cdiv(Nn, WPB), TB, 0, stream>>>(x2, batch, pm, pxu, cnt, Nn, HC);
  k_pool_final<<<cdiv((long)G * 2 * HC, TB), TB, 0, stream>>>(pm, pxu, cnt, hcat, G, HC);

  // 7) MLP head
  {
    dim3 g1(cdiv(FC1, 128), cdiv(G, 64));
    k_wmma_gemm<<<g1, TB, 0, stream>>>(hcat, W_fc1, b_fc1, h1, G, FC1, 2 * HC, 1);
    dim3 g2(cdiv(GO, 128), cdiv(G, 64));
    k_wmma_gemm<<<g2, TB, 0, stream>>>(h1, W_go, b_go, h2, G, GO, FC1, 0);
    dim3 g3(cdiv(SEMI, 128), cdiv(G, 64));
    k_wmma_gemm<<<g3, TB, 0, stream>>>(h2, W_semi, b_semi, h3, G, SEMI, GO, 0);
  }
  k_final_dot<<<cdiv(G, WPB), TB, 0, stream>>>(h3, W_fin, out, G, SEMI);
}